// QATLinear_81724637709038
// MI455X (gfx1250) — compile-verified
//
#include <hip/hip_runtime.h>

typedef __attribute__((ext_vector_type(2))) float v2f;
typedef __attribute__((ext_vector_type(4))) float v4f;
typedef __attribute__((ext_vector_type(8))) float v8f;

#define TOKENS 8192
#define D_IN   4096
#define D_OUT  16384

#define BM 128
#define BN 128
#define BK 32
#define LDSTR 36   // padded row stride in floats: 144B rows -> 16B aligned, conflict-free 16-lane frag reads

__global__ __launch_bounds__(256) void qat_linear_wmma_f32(
    const float* __restrict__ X,   // [TOKENS, D_IN]
    const float* __restrict__ W,   // [D_OUT, D_IN]
    const float* __restrict__ S,   // [D_OUT]
    const float* __restrict__ Bv,  // [D_OUT]
    float* __restrict__ OUT)       // [TOKENS, D_OUT]
{
    __shared__ float Alds[BM * LDSTR];   // [m][k] tile of x
    __shared__ float Blds[BN * LDSTR];   // [n][k] tile of sign(w)

    const int tid  = threadIdx.x;
    const int lane = tid & 31;
    const int wave = tid >> 5;    // 0..7
    const int wm   = wave & 1;    // M slab: 0..1  (64 rows each)
    const int wn   = wave >> 1;   // N slab: 0..3  (32 cols each)
    const int l15  = lane & 15;
    const int hi   = lane >> 4;   // 0 or 1 (K half for A/B frags, M half for C/D)

    const int m0 = blockIdx.x * BM;   // x-fastest over M: concurrent blocks share the weight slab in L2
    const int n0 = blockIdx.y * BN;

    const v8f vzero = {0.f,0.f,0.f,0.f,0.f,0.f,0.f,0.f};
    v8f acc[4][2];
#pragma unroll
    for (int mt = 0; mt < 4; ++mt)
#pragma unroll
        for (int nt = 0; nt < 2; ++nt)
            acc[mt][nt] = vzero;

    for (int kk = 0; kk < D_IN; kk += BK) {
        // ---- stage tiles into LDS: 1024 float4 chunks, 4 per thread ----
#pragma unroll
        for (int i = 0; i < 4; ++i) {
            const int c   = tid + i * 256;     // 0..1023
            const int row = c >> 3;            // 0..127
            const int col = (c & 7) << 2;      // 0,4,...,28

            // A: straight copy of x
            const v4f* xp = (const v4f*)(X + (size_t)(m0 + row) * D_IN + kk + col);
            v4f av = *xp;
            *(v4f*)(&Alds[row * LDSTR + col]) = av;

            // B: sign(w) as exact +-1.0f (zeros map to +1), non-temporal (streamed once)
            const v4f* wp = (const v4f*)(W + (size_t)(n0 + row) * D_IN + kk + col);
            v4f wv = __builtin_nontemporal_load(wp);
            v4f sv;
            sv.x = (wv.x >= 0.f) ? 1.f : -1.f;
            sv.y = (wv.y >= 0.f) ? 1.f : -1.f;
            sv.z = (wv.z >= 0.f) ? 1.f : -1.f;
            sv.w = (wv.w >= 0.f) ? 1.f : -1.f;
            *(v4f*)(&Blds[row * LDSTR + col]) = sv;

            // prefetch next K slab (global_prefetch_b8)
            if (kk + BK < D_IN) {
                __builtin_prefetch(X + (size_t)(m0 + row) * D_IN + kk + BK + col, 0, 1);
                __builtin_prefetch(W + (size_t)(n0 + row) * D_IN + kk + BK + col, 0, 1);
            }
        }
        __syncthreads();

        // ---- 8 K-steps of 4, 8 WMMAs each ----
#pragma unroll
        for (int k2 = 0; k2 < BK; k2 += 4) {
            v2f afrag[4], bfrag[2];
#pragma unroll
            for (int mt = 0; mt < 4; ++mt) {
                const int row = wm * 64 + mt * 16 + l15;
                afrag[mt] = *(const v2f*)(&Alds[row * LDSTR + k2 + 2 * hi]);
            }
#pragma unroll
            for (int nt = 0; nt < 2; ++nt) {
                const int row = wn * 32 + nt * 16 + l15;
                bfrag[nt] = *(const v2f*)(&Blds[row * LDSTR + k2 + 2 * hi]);
            }
#pragma unroll
            for (int mt = 0; mt < 4; ++mt)
#pragma unroll
                for (int nt = 0; nt < 2; ++nt)
                    acc[mt][nt] = __builtin_amdgcn_wmma_f32_16x16x4_f32(
                        /*neg_a=*/false, afrag[mt],
                        /*neg_b=*/false, bfrag[nt],
                        /*c_mod=*/(short)0, acc[mt][nt],
                        /*reuse_a=*/false, /*reuse_b=*/false);
        }
        __syncthreads();
    }

    // ---- epilogue: out = acc*scale[n] + b[n]; C/D layout: col = l15, row = vgpr + 8*hi ----
#pragma unroll
    for (int nt = 0; nt < 2; ++nt) {
        const int ncol = n0 + wn * 32 + nt * 16 + l15;
        const float sc = S[ncol];
        const float bi = Bv[ncol];
#pragma unroll
        for (int mt = 0; mt < 4; ++mt) {
            const int mbase = m0 + wm * 64 + mt * 16 + hi * 8;
#pragma unroll
            for (int v = 0; v < 8; ++v) {
                float r = fmaf(acc[mt][nt][v], sc, bi);
                __builtin_nontemporal_store(r, OUT + (size_t)(mbase + v) * D_OUT + ncol);
            }
        }
    }
}

extern "C" void kernel_launch(void* const* d_in, const int* in_sizes, int n_in,
                              void* d_out, int out_size, void* d_ws, size_t ws_size,
                              hipStream_t stream) {
    const float* x = (const float*)d_in[0];
    const float* w = (const float*)d_in[1];
    const float* s = (const float*)d_in[2];
    const float* b = (const float*)d_in[3];
    float* out = (float*)d_out;

    dim3 grid(TOKENS / BM, D_OUT / BN);   // (64, 128), x-fastest over M
    qat_linear_wmma_f32<<<grid, 256, 0, stream>>>(x, w, s, b, out);
}